// RNNModel_45561013076453
// MI455X (gfx1250) — compile-verified
//
#include <hip/hip_runtime.h>
#include <math.h>

#define T_STEPS 128
#define BATCH   64
#define NTOKEN  10000
#define NB      6
#define BS      100
#define NHID    600   // NB*BS
#define DK_INP  128
#define MHA_E   64    // NH_MHA*DK_MHA
#define TOPK    4

typedef __attribute__((ext_vector_type(2))) float v2f;
typedef __attribute__((ext_vector_type(8))) float v8f;

__device__ __forceinline__ float sigmoidf_(float x) { return 1.0f / (1.0f + __expf(-x)); }

// ---------------- init: copy hidden -> hx workspace ----------------
__global__ void init_hx_kernel(const float* __restrict__ hidden, float* __restrict__ hx) {
    int i = blockIdx.x * blockDim.x + threadIdx.x;
    if (i < BATCH * NHID) hx[i] = hidden[i];
}

// ---------------- one recurrent step, one block per batch element ----------------
__global__ __launch_bounds__(384)
void rim_step_kernel(const int* __restrict__ input,
                     const float* __restrict__ encoder_w,
                     const float* __restrict__ inp_wq,   // (NB,BS,128)
                     const float* __restrict__ inp_wk,   // (2,600,128) row0 used
                     const float* __restrict__ inp_wv,   // (2,600,128) row0 used
                     const float* __restrict__ inp_fc_w, // (100,128)
                     const float* __restrict__ inp_fc_b, // (100)
                     const float* __restrict__ mha_wq,   // (NB,100,64)
                     const float* __restrict__ mha_wk,
                     const float* __restrict__ mha_wv,
                     const float* __restrict__ mha_fc_w, // (100,64)
                     const float* __restrict__ mha_fc_b,
                     const float* __restrict__ mha_gate_w,
                     const float* __restrict__ mha_gate_b,
                     const float* __restrict__ gru_w_ih, // (NB,300,100)
                     const float* __restrict__ gru_w_hh,
                     const float* __restrict__ gru_b_ih, // (NB,300)
                     const float* __restrict__ gru_b_hh,
                     float* __restrict__ hx_g,           // (B,600) state
                     float* __restrict__ outputs,        // (T,B,600)
                     int t)
{
    const int tid = threadIdx.x;
    const int b   = blockIdx.x;

    __shared__ float xrow[NHID];
    __shared__ float hxs[NHID];
    __shared__ float k0[DK_INP];
    __shared__ float v0[DK_INP];
    __shared__ float fcv0[BS];
    __shared__ float red[384];
    __shared__ float sc[NB];
    __shared__ float msk[NB];
    __shared__ float iu[NHID];
    __shared__ float gx[NB * 3 * BS];
    __shared__ float gh[NB * 3 * BS];
    __shared__ float hn[NHID];
    __shared__ float qs[NB * MHA_E];
    __shared__ float ks[NB * MHA_E];
    __shared__ float vs[NB * MHA_E];
    __shared__ float attw[4 * NB * NB];
    __shared__ float outs[NB * MHA_E];

    // load embedding row (gather) and current hidden state
    const int token = input[t * BATCH + b];
    for (int i = tid; i < NHID; i += 384) {
        xrow[i] = encoder_w[token * NHID + i];
        hxs[i]  = hx_g[b * NHID + i];
    }
    __syncthreads();

    // k0 = x @ Wk[0], v0 = x @ Wv[0]   (128 each)
    if (tid < DK_INP) {
        float a = 0.f;
        for (int d = 0; d < NHID; ++d) a += xrow[d] * inp_wk[d * DK_INP + tid];
        k0[tid] = a;
    } else if (tid < 2 * DK_INP) {
        const int e = tid - DK_INP;
        float a = 0.f;
        for (int d = 0; d < NHID; ++d) a += xrow[d] * inp_wv[d * DK_INP + e];
        v0[e] = a;
    }
    __syncthreads();

    // fcv0 = v0 @ fc_w^T  (100)
    if (tid < BS) {
        float a = 0.f;
        for (int j = 0; j < DK_INP; ++j) a += v0[j] * inp_fc_w[tid * DK_INP + j];
        fcv0[tid] = a;
    }

    // attention scores: s_n = (hx_n @ Wq_n) . k0 ; attn0 = sigmoid(s*scale)
    {
        const int n = tid >> 6;     // 0..5
        const int i = tid & 63;     // 0..63
        float acc = 0.f;
        for (int eo = 0; eo < 2; ++eo) {
            const int e = i + 64 * eo;
            float qp = 0.f;
            const float* w = inp_wq + n * (BS * DK_INP) + e;
            for (int d = 0; d < BS; ++d) qp += hxs[n * BS + d] * w[d * DK_INP];
            acc += qp * k0[e];
        }
        red[tid] = acc;
    }
    __syncthreads();
    for (int s = 32; s > 0; s >>= 1) {
        if ((tid & 63) < s) red[tid] += red[tid + s];
        __syncthreads();
    }
    if ((tid & 63) == 0) {
        const float scale = 0.08838834764831845f; // 1/sqrt(128)
        sc[tid >> 6] = sigmoidf_(red[tid] * scale);
    }
    __syncthreads();

    // top-k (k=4 of 6) mask, thread 0
    if (tid == 0) {
        float v[NB]; bool used[NB];
        for (int n = 0; n < NB; ++n) { v[n] = sc[n]; used[n] = false; }
        float kth = 0.f;
        for (int it = 0; it < TOPK; ++it) {
            int bi = -1; float bv = -1e30f;
            for (int n = 0; n < NB; ++n)
                if (!used[n] && v[n] > bv) { bv = v[n]; bi = n; }
            used[bi] = true; kth = bv;
        }
        for (int n = 0; n < NB; ++n) msk[n] = (sc[n] > kth - 0.01f) ? 1.f : 0.f;
    }

    // inp_use = attn0 * fcv0 + fc_b
    for (int idx = tid; idx < NHID; idx += 384) {
        const int n = idx / BS, c = idx % BS;
        iu[idx] = sc[n] * fcv0[c] + inp_fc_b[c];
    }
    __syncthreads();

    // BlockGRU gate pre-activations
    for (int idx = tid; idx < NB * 3 * BS; idx += 384) {
        const int n = idx / (3 * BS), g = idx % (3 * BS);
        float a1 = gru_b_ih[n * 3 * BS + g];
        float a2 = gru_b_hh[n * 3 * BS + g];
        const float* wi = gru_w_ih + n * (3 * BS * BS) + g * BS;
        const float* wh = gru_w_hh + n * (3 * BS * BS) + g * BS;
        for (int d = 0; d < BS; ++d) {
            a1 += iu[n * BS + d]  * wi[d];
            a2 += hxs[n * BS + d] * wh[d];
        }
        gx[idx] = a1; gh[idx] = a2;
    }
    __syncthreads();

    // GRU combine (PyTorch order r,z,n)
    for (int idx = tid; idx < NHID; idx += 384) {
        const int n = idx / BS, c = idx % BS;
        const int g0 = n * 3 * BS;
        const float r  = sigmoidf_(gx[g0 + c]          + gh[g0 + c]);
        const float z  = sigmoidf_(gx[g0 + BS + c]     + gh[g0 + BS + c]);
        const float nn = tanhf(gx[g0 + 2 * BS + c] + r * gh[g0 + 2 * BS + c]);
        hn[idx] = (1.f - z) * nn + z * hxs[idx];
    }
    __syncthreads();

    // MHA q/k/v projections (NB x 64 each)
    {
        const int n = tid >> 6, e = tid & 63;
        float aq = 0.f, ak = 0.f, av = 0.f;
        const float* wq = mha_wq + n * (BS * MHA_E) + e;
        const float* wk = mha_wk + n * (BS * MHA_E) + e;
        const float* wv = mha_wv + n * (BS * MHA_E) + e;
        for (int d = 0; d < BS; ++d) {
            const float hv = hn[n * BS + d];
            aq += hv * wq[d * MHA_E];
            ak += hv * wk[d * MHA_E];
            av += hv * wv[d * MHA_E];
        }
        qs[tid] = aq; ks[tid] = ak; vs[tid] = av;
    }
    __syncthreads();

    // attention softmax rows: 4 heads x 6 queries
    if (tid < 24) {
        const int h = tid / NB, i = tid % NB;
        float s[NB];
        float mx = -1e30f;
        for (int j = 0; j < NB; ++j) {
            float a = 0.f;
            for (int d = 0; d < 16; ++d)
                a += qs[i * MHA_E + h * 16 + d] * ks[j * MHA_E + h * 16 + d];
            s[j] = a * 0.25f; // 1/sqrt(16)
            mx = fmaxf(mx, s[j]);
        }
        float sum = 0.f;
        for (int j = 0; j < NB; ++j) { s[j] = __expf(s[j] - mx); sum += s[j]; }
        const float inv = 1.f / sum;
        for (int j = 0; j < NB; ++j) attw[(h * NB + i) * NB + j] = s[j] * inv;
    }
    __syncthreads();

    // out = attn @ v
    {
        const int n = tid >> 6, e = tid & 63, h = e >> 4;
        float o = 0.f;
        for (int j = 0; j < NB; ++j) o += attw[(h * NB + n) * NB + j] * vs[j * MHA_E + e];
        outs[tid] = o;
    }
    __syncthreads();

    // gated projection + masked state update
    for (int idx = tid; idx < NHID; idx += 384) {
        const int n = idx / BS, c = idx % BS;
        float p = mha_fc_b[c];
        float g = mha_gate_b[c];
        for (int e = 0; e < MHA_E; ++e) {
            const float o = outs[n * MHA_E + e];
            p += o * mha_fc_w[c * MHA_E + e];
            g += o * mha_gate_w[c * MHA_E + e];
        }
        const float hatt = sigmoidf_(g) * tanhf(p);
        const float m = msk[n];
        const float nv = m * hatt + (1.f - m) * hxs[idx];
        hx_g[b * NHID + idx] = nv;
        outputs[(size_t)t * BATCH * NHID + b * NHID + idx] = nv;
    }
}

// ---------------- decoder GEMM with V_WMMA_F32_16X16X4_F32 ----------------
// D(8192 x 10000) = outputs(8192 x 600) @ dec_w^T + dec_b
// block = 128 threads (4 waves); block tile 128(M) x 64(N); wave tile 32(M) x 64(N)
// software-pipelined: fragments for k+4 are loaded before the WMMAs for k issue,
// so global loads overlap the matrix pipe instead of s_wait_loadcnt 0 per WMMA.
__global__ __launch_bounds__(128)
void decode_wmma_kernel(const float* __restrict__ A,      // (8192,600)
                        const float* __restrict__ dec_w,  // (10000,600)
                        const float* __restrict__ dec_b,  // (10000)
                        float* __restrict__ D)            // (8192,10000)
{
    const int lane  = threadIdx.x & 31;
    const int wave  = threadIdx.x >> 5;
    const int n0    = blockIdx.x * 64;              // N group start
    const int m0    = blockIdx.y * 128 + wave * 32; // this wave's 32-row M tile

    const int lmod  = lane & 15;
    const int khalf = (lane >> 4) << 1;             // 0 or 2

    v8f c00 = {}, c01 = {}, c02 = {}, c03 = {};     // M tile 0 x 4 N tiles
    v8f c10 = {}, c11 = {}, c12 = {}, c13 = {};     // M tile 1 x 4 N tiles

    const int ncol0 = n0 + lmod;
    const int ncol1 = n0 + 16 + lmod;
    const int ncol2 = n0 + 32 + lmod;
    const int ncol3 = n0 + 48 + lmod;
    const bool p0 = ncol0 < NTOKEN, p1 = ncol1 < NTOKEN, p2 = ncol2 < NTOKEN, p3 = ncol3 < NTOKEN;

    const float* a0p = A + (size_t)(m0 + lmod) * NHID + khalf;
    const float* a1p = A + (size_t)(m0 + 16 + lmod) * NHID + khalf;
    const float* b0p = p0 ? dec_w + (size_t)ncol0 * NHID + khalf : dec_w + khalf;
    const float* b1p = p1 ? dec_w + (size_t)ncol1 * NHID + khalf : dec_w + khalf;
    const float* b2p = p2 ? dec_w + (size_t)ncol2 * NHID + khalf : dec_w + khalf;
    const float* b3p = p3 ? dec_w + (size_t)ncol3 * NHID + khalf : dec_w + khalf;

    // prologue: fragments for k = 0
    v2f a0c = *reinterpret_cast<const v2f*>(a0p);
    v2f a1c = *reinterpret_cast<const v2f*>(a1p);
    v2f b0c = *reinterpret_cast<const v2f*>(b0p);
    v2f b1c = *reinterpret_cast<const v2f*>(b1p);
    v2f b2c = *reinterpret_cast<const v2f*>(b2p);
    v2f b3c = *reinterpret_cast<const v2f*>(b3p);

    for (int k = 4; k < NHID; k += 4) {
        // prefetch next iteration's fragments (overlaps WMMAs below)
        v2f a0n = *reinterpret_cast<const v2f*>(a0p + k);
        v2f a1n = *reinterpret_cast<const v2f*>(a1p + k);
        v2f b0n = *reinterpret_cast<const v2f*>(b0p + k);
        v2f b1n = *reinterpret_cast<const v2f*>(b1p + k);
        v2f b2n = *reinterpret_cast<const v2f*>(b2p + k);
        v2f b3n = *reinterpret_cast<const v2f*>(b3p + k);

        c00 = __builtin_amdgcn_wmma_f32_16x16x4_f32(false, a0c, false, b0c, (short)0, c00, false, false);
        c01 = __builtin_amdgcn_wmma_f32_16x16x4_f32(false, a0c, false, b1c, (short)0, c01, false, false);
        c02 = __builtin_amdgcn_wmma_f32_16x16x4_f32(false, a0c, false, b2c, (short)0, c02, false, false);
        c03 = __builtin_amdgcn_wmma_f32_16x16x4_f32(false, a0c, false, b3c, (short)0, c03, false, false);
        c10 = __builtin_amdgcn_wmma_f32_16x16x4_f32(false, a1c, false, b0c, (short)0, c10, false, false);
        c11 = __builtin_amdgcn_wmma_f32_16x16x4_f32(false, a1c, false, b1c, (short)0, c11, false, false);
        c12 = __builtin_amdgcn_wmma_f32_16x16x4_f32(false, a1c, false, b2c, (short)0, c12, false, false);
        c13 = __builtin_amdgcn_wmma_f32_16x16x4_f32(false, a1c, false, b3c, (short)0, c13, false, false);

        a0c = a0n; a1c = a1n;
        b0c = b0n; b1c = b1n; b2c = b2n; b3c = b3n;
    }
    // epilogue: last k-step
    c00 = __builtin_amdgcn_wmma_f32_16x16x4_f32(false, a0c, false, b0c, (short)0, c00, false, false);
    c01 = __builtin_amdgcn_wmma_f32_16x16x4_f32(false, a0c, false, b1c, (short)0, c01, false, false);
    c02 = __builtin_amdgcn_wmma_f32_16x16x4_f32(false, a0c, false, b2c, (short)0, c02, false, false);
    c03 = __builtin_amdgcn_wmma_f32_16x16x4_f32(false, a0c, false, b3c, (short)0, c03, false, false);
    c10 = __builtin_amdgcn_wmma_f32_16x16x4_f32(false, a1c, false, b0c, (short)0, c10, false, false);
    c11 = __builtin_amdgcn_wmma_f32_16x16x4_f32(false, a1c, false, b1c, (short)0, c11, false, false);
    c12 = __builtin_amdgcn_wmma_f32_16x16x4_f32(false, a1c, false, b2c, (short)0, c12, false, false);
    c13 = __builtin_amdgcn_wmma_f32_16x16x4_f32(false, a1c, false, b3c, (short)0, c13, false, false);

    // biases
    const float bb0 = p0 ? dec_b[ncol0] : 0.f;
    const float bb1 = p1 ? dec_b[ncol1] : 0.f;
    const float bb2 = p2 ? dec_b[ncol2] : 0.f;
    const float bb3 = p3 ? dec_b[ncol3] : 0.f;

    // store: C/D layout — VGPR j, lane L: M = j + (L>=16 ? 8 : 0), N = L%16
    const int mbase0 = m0 + ((lane >> 4) << 3);        // M tile 0
    const int mbase1 = m0 + 16 + ((lane >> 4) << 3);   // M tile 1
    #pragma unroll
    for (int j = 0; j < 8; ++j) {
        const size_t r0 = (size_t)(mbase0 + j) * NTOKEN;
        const size_t r1 = (size_t)(mbase1 + j) * NTOKEN;
        if (p0) { D[r0 + ncol0] = c00[j] + bb0; D[r1 + ncol0] = c10[j] + bb0; }
        if (p1) { D[r0 + ncol1] = c01[j] + bb1; D[r1 + ncol1] = c11[j] + bb1; }
        if (p2) { D[r0 + ncol2] = c02[j] + bb2; D[r1 + ncol2] = c12[j] + bb2; }
        if (p3) { D[r0 + ncol3] = c03[j] + bb3; D[r1 + ncol3] = c13[j] + bb3; }
    }
}

// ---------------- tail: hx_final + scalar ----------------
__global__ void tail_kernel(const float* __restrict__ hx, float* __restrict__ out) {
    int i = blockIdx.x * blockDim.x + threadIdx.x;
    const size_t base = (size_t)T_STEPS * BATCH * NTOKEN;
    if (i < BATCH * NHID) out[base + i] = hx[i];
    if (i == BATCH * NHID) out[base + BATCH * NHID] = 0.0f;
}

extern "C" void kernel_launch(void* const* d_in, const int* in_sizes, int n_in,
                              void* d_out, int out_size, void* d_ws, size_t ws_size,
                              hipStream_t stream) {
    const int*   input      = (const int*)  d_in[0];
    const float* hidden     = (const float*)d_in[1];
    const float* encoder_w  = (const float*)d_in[2];
    const float* inp_wq     = (const float*)d_in[3];
    const float* inp_wk     = (const float*)d_in[4];
    const float* inp_wv     = (const float*)d_in[5];
    const float* inp_fc_w   = (const float*)d_in[6];
    const float* inp_fc_b   = (const float*)d_in[7];
    const float* mha_wq     = (const float*)d_in[8];
    const float* mha_wk     = (const float*)d_in[9];
    const float* mha_wv     = (const float*)d_in[10];
    const float* mha_fc_w   = (const float*)d_in[11];
    const float* mha_fc_b   = (const float*)d_in[12];
    const float* mha_gate_w = (const float*)d_in[13];
    const float* mha_gate_b = (const float*)d_in[14];
    const float* gru_w_ih   = (const float*)d_in[15];
    const float* gru_w_hh   = (const float*)d_in[16];
    const float* gru_b_ih   = (const float*)d_in[17];
    const float* gru_b_hh   = (const float*)d_in[18];
    const float* dec_w      = (const float*)d_in[19];
    const float* dec_b      = (const float*)d_in[20];

    float* out = (float*)d_out;
    float* ws  = (float*)d_ws;
    float* hx       = ws;                   // B*NHID
    float* outputs  = ws + BATCH * NHID;    // T*B*NHID

    init_hx_kernel<<<(BATCH * NHID + 255) / 256, 256, 0, stream>>>(hidden, hx);

    for (int t = 0; t < T_STEPS; ++t) {
        rim_step_kernel<<<BATCH, 384, 0, stream>>>(
            input, encoder_w, inp_wq, inp_wk, inp_wv, inp_fc_w, inp_fc_b,
            mha_wq, mha_wk, mha_wv, mha_fc_w, mha_fc_b, mha_gate_w, mha_gate_b,
            gru_w_ih, gru_w_hh, gru_b_ih, gru_b_hh, hx, outputs, t);
    }

    dim3 grid((NTOKEN + 63) / 64, (T_STEPS * BATCH) / 128);
    decode_wmma_kernel<<<grid, 128, 0, stream>>>(outputs, dec_w, dec_b, out);

    tail_kernel<<<(BATCH * NHID + 256) / 256, 256, 0, stream>>>(hx, out);
}